// GCN_37357625541048
// MI455X (gfx1250) — compile-verified
//
#include <hip/hip_runtime.h>
#include <math.h>

typedef __attribute__((ext_vector_type(2))) float v2f;
typedef __attribute__((ext_vector_type(8))) float v8f;

#define N_NODES 100000
#define N_EDGES 1600000

// ---------------- degree / normalization ----------------

__global__ void k_deg_init(float* __restrict__ deg, int n) {
  int i = blockIdx.x * blockDim.x + threadIdx.x;
  if (i < n) deg[i] = 1.0f;  // self-loop contribution
}

__global__ void k_deg_edges(const int* __restrict__ dst, float* __restrict__ deg, int e) {
  int i = blockIdx.x * blockDim.x + threadIdx.x;
  if (i < e) atomicAdd(&deg[dst[i]], 1.0f);
}

__global__ void k_deg_rsqrt(float* __restrict__ deg, int n) {
  int i = blockIdx.x * blockDim.x + threadIdx.x;
  if (i < n) deg[i] = rsqrtf(deg[i]);  // deg >= 1 always (self-loops)
}

// ---------------- fused WMMA fp32 GEMM + aggregation init ----------------
// H[n,Fout] = act(X)[n,Fin] @ W[Fin,Fout]       (act = relu if relu_in)
// AGG[i,f]  = bias[f] + H[i,f] * dis[i]^2       (self-loop term folded in)
//
// Block = 256 threads (8 waves), computes a 64-row x 128-col output tile.
// Wave w: row-tile rt = w&3 (16 rows), col-tiles (w>>2)*64 + {0,16,32,48}.
// K is staged in LDS chunks of 32 with zero padding, so the inner loop is
// branch-free: ds_load_b64 (A) + 4 x (ds_load_b64 (B) + v_wmma_f32_16x16x4_f32).
//
// VGPR layouts per CDNA5 ISA:
//   A 16x4:  m=lane&15, half=lane>>4 -> float2 = A[m][k0+2*half .. +1]
//   B 4x16:  col=lane&15             -> float2 = W[k0+2*half][col], W[k0+2*half+1][col]
//   C 16x16: VGPR r -> row r + 8*half, col = lane&15
#define KC 32
__launch_bounds__(256)
__global__ void k_gemm_wmma(const float* __restrict__ X, int Fin,
                            const float* __restrict__ Wg, int Fout,
                            const float* __restrict__ bias,
                            const float* __restrict__ dis, int relu_in,
                            float* __restrict__ H, float* __restrict__ AGG, int n) {
  __shared__ float As[64][36];          // stride 36: conflict-free b64 reads
  __shared__ float Ws[(KC / 2) * 288];  // pair layout, stride 288: halves split banks

  const int row0 = blockIdx.x * 64;
  const int tid  = threadIdx.x;
  const int wave = tid >> 5;
  const int lane = tid & 31;
  const int m    = lane & 15;
  const int half = lane >> 4;
  const int rt   = wave & 3;           // row tile 0..3
  const int cbase = (wave >> 2) * 64;  // col group base: 0 or 64

  v8f acc[4] = {v8f{}, v8f{}, v8f{}, v8f{}};

  const int nchunks = (Fin + KC - 1) / KC;
  for (int kc = 0; kc < nchunks; ++kc) {
    const int kc0 = kc * KC;
    __syncthreads();
    // stage A chunk: 64 rows x 32 k (zero-padded, optional relu)
    #pragma unroll
    for (int idx = tid; idx < 64 * KC; idx += 256) {
      int r = idx >> 5, kk = idx & (KC - 1);
      int gk = kc0 + kk, row = row0 + r;
      float v = (row < n && gk < Fin) ? X[(long)row * Fin + gk] : 0.0f;
      if (relu_in) v = fmaxf(v, 0.0f);
      As[r][kk] = v;
    }
    // stage W chunk in B pair layout: Ws[(kk>>1)*288 + col*2 + (kk&1)]
    #pragma unroll
    for (int idx = tid; idx < KC * 128; idx += 256) {
      int col = idx & 127, kk = idx >> 7;
      int gk = kc0 + kk;
      float v = (gk < Fin && col < Fout) ? Wg[(long)gk * Fout + col] : 0.0f;
      Ws[(kk >> 1) * 288 + col * 2 + (kk & 1)] = v;
    }
    __syncthreads();

    const int ar = rt * 16 + m;
    #pragma unroll
    for (int k0 = 0; k0 < KC; k0 += 4) {
      v2f a = *(const v2f*)&As[ar][k0 + 2 * half];
      #pragma unroll
      for (int j = 0; j < 4; ++j) {
        int col = cbase + j * 16 + m;
        v2f b = *(const v2f*)&Ws[((k0 >> 1) + half) * 288 + col * 2];
        acc[j] = __builtin_amdgcn_wmma_f32_16x16x4_f32(false, a, false, b,
                                                       (short)0, acc[j], false, false);
      }
    }
  }

  // epilogue: write H and bias/self-loop-initialized AGG
  #pragma unroll
  for (int j = 0; j < 4; ++j) {
    int col = cbase + j * 16 + m;
    if (col >= Fout) continue;
    float bv = bias[col];
    #pragma unroll
    for (int r = 0; r < 8; ++r) {
      int row = row0 + rt * 16 + r + 8 * half;
      if (row < n) {
        float dv = dis[row];
        long o = (long)row * Fout + col;
        float h = acc[j][r];
        H[o]   = h;
        AGG[o] = bv + h * dv * dv;
      }
    }
  }
}

// ---------------- edge scatter: agg[dst,:] += H[src,:] * dis[src]*dis[dst] ----------------
__launch_bounds__(256)
__global__ void k_scatter(const float* __restrict__ H, const float* __restrict__ dis,
                          const int* __restrict__ src, const int* __restrict__ dst,
                          float* __restrict__ agg, int E, int F) {
  int e = blockIdx.x * (blockDim.x >> 5) + (threadIdx.x >> 5);
  if (e >= E) return;
  int lane = threadIdx.x & 31;
  int s = src[e], d = dst[e];
  float nrm = dis[s] * dis[d];
  const float* hs = H + (long)s * F;
  float* ad = agg + (long)d * F;
  for (int f = lane; f < F; f += 32)
    atomicAdd(&ad[f], hs[f] * nrm);
}

// ---------------- final activation ----------------
__global__ void k_sigmoid(const float* __restrict__ in, float* __restrict__ out, long total) {
  long i = (long)blockIdx.x * blockDim.x + threadIdx.x;
  if (i < total) out[i] = 1.0f / (1.0f + __expf(-in[i]));
}

// ---------------- launcher ----------------

extern "C" void kernel_launch(void* const* d_in, const int* in_sizes, int n_in,
                              void* d_out, int out_size, void* d_ws, size_t ws_size,
                              hipStream_t stream) {
  (void)in_sizes; (void)n_in; (void)out_size; (void)ws_size;
  const float* x    = (const float*)d_in[0];           // [N,50]
  const int*   src  = (const int*)d_in[1];             // edge_index[0]
  const int*   dst  = ((const int*)d_in[1]) + N_EDGES; // edge_index[1]
  const float* W1 = (const float*)d_in[2]; const float* b1 = (const float*)d_in[3];
  const float* W2 = (const float*)d_in[4]; const float* b2 = (const float*)d_in[5];
  const float* W3 = (const float*)d_in[6]; const float* b3 = (const float*)d_in[7];
  const float* W4 = (const float*)d_in[8]; const float* b4 = (const float*)d_in[9];
  float* out = (float*)d_out;                          // [N,121]

  // workspace layout (fp32)
  float* ws   = (float*)d_ws;
  float* dis  = ws;                                    // N (deg -> dis in place)
  float* bufH = ws + 102400;                           // N*128
  float* agg0 = bufH + (long)N_NODES * 128;            // N*128
  float* agg1 = agg0 + (long)N_NODES * 128;            // N*128

  const int TB = 256;
  const int nb_nodes = (N_NODES + TB - 1) / TB;
  const int nb_edges = (N_EDGES + TB - 1) / TB;
  const int nb_gemm  = (N_NODES + 63) / 64;
  const int nb_scat  = (N_EDGES + 7) / 8;              // 8 waves/block, 1 wave/edge

  // degrees -> dis
  k_deg_init <<<nb_nodes, TB, 0, stream>>>(dis, N_NODES);
  k_deg_edges<<<nb_edges, TB, 0, stream>>>(dst, dis, N_EDGES);
  k_deg_rsqrt<<<nb_nodes, TB, 0, stream>>>(dis, N_NODES);

  // Layer 1: x[N,50] @ W1 -> 128
  k_gemm_wmma<<<nb_gemm, 256, 0, stream>>>(x, 50, W1, 128, b1, dis, 0, bufH, agg0, N_NODES);
  k_scatter  <<<nb_scat, TB, 0, stream>>>(bufH, dis, src, dst, agg0, N_EDGES, 128);

  // Layer 2 (relu applied while staging A)
  k_gemm_wmma<<<nb_gemm, 256, 0, stream>>>(agg0, 128, W2, 128, b2, dis, 1, bufH, agg1, N_NODES);
  k_scatter  <<<nb_scat, TB, 0, stream>>>(bufH, dis, src, dst, agg1, N_EDGES, 128);

  // Layer 3
  k_gemm_wmma<<<nb_gemm, 256, 0, stream>>>(agg1, 128, W3, 128, b3, dis, 1, bufH, agg0, N_NODES);
  k_scatter  <<<nb_scat, TB, 0, stream>>>(bufH, dis, src, dst, agg0, N_EDGES, 128);

  // Layer 4: -> 121, sigmoid into d_out
  k_gemm_wmma<<<nb_gemm, 256, 0, stream>>>(agg0, 128, W4, 121, b4, dis, 1, bufH, agg1, N_NODES);
  k_scatter  <<<nb_scat, TB, 0, stream>>>(bufH, dis, src, dst, agg1, N_EDGES, 121);

  const long tot121 = (long)N_NODES * 121;
  k_sigmoid  <<<(int)((tot121 + TB - 1) / TB), TB, 0, stream>>>(agg1, out, tot121);
}